// LSTMBehaviorModel_87582973100043
// MI455X (gfx1250) — compile-verified
//
#include <hip/hip_runtime.h>
#include <hip/hip_bf16.h>

typedef float v2f __attribute__((ext_vector_type(2)));
typedef float v8f __attribute__((ext_vector_type(8)));

#define S_LEN 256
#define F_IN  5
#define HID   64
#define GATES 256
#define BT    16      // batch rows per workgroup (one WMMA M tile)
#define NTHREADS 256  // 8 waves: waves 0-3 = layer0, waves 4-7 = layer1 (pipelined)

// ---- LDS layout (float offsets) ----
#define OFF_WHH0 0
#define OFF_WIH1 (OFF_WHH0 + GATES*HID)
#define OFF_WHH1 (OFF_WIH1 + GATES*HID)
#define OFF_WIH0 (OFF_WHH1 + GATES*HID)   // padded [256][8]
#define OFF_B0   (OFF_WIH0 + GATES*8)
#define OFF_B1   (OFF_B0 + GATES)
#define OFF_H0   (OFF_B1 + GATES)
#define OFF_H1   (OFF_H0 + BT*HID)
#define OFF_X    (OFF_H1 + BT*HID)        // padded [16][8]
#define SMEM_FLOATS (OFF_X + BT*8)        // 53888 floats = 210.5 KB (< 320 KB WGP LDS)

__device__ __forceinline__ float sigf(float x) { return 1.f / (1.f + __expf(-x)); }
__device__ __forceinline__ float tanh_f(float x) {
  float e = __expf(-2.f * fabsf(x));
  float r = (1.f - e) / (1.f + e);
  return copysignf(r, x);
}

__device__ __forceinline__ v8f wmma4(v2f a, v2f b, v8f c) {
  // D = A(16x4,f32) * B(4x16,f32) + C(16x16,f32)
  return __builtin_amdgcn_wmma_f32_16x16x4_f32(false, a, false, b, (short)0, c, false, false);
}

// splat fused bias for this wave's 4 gate tiles into the accumulators
__device__ __forceinline__ void acc_init(v8f acc[4], const float* sm, int boff, int col) {
#pragma unroll
  for (int g = 0; g < 4; ++g) {
    float b = sm[boff + g * 64 + col];
#pragma unroll
    for (int r = 0; r < 8; ++r) acc[g][r] = b;
  }
}

// acc[g] += A(16x64 @ sm[aoff], rows m) * B(64x16 cols g*64+col of weight [256][64] @ sm[woff])
__device__ __forceinline__ void gemm_k64(v8f acc[4], const float* sm,
                                         int aoff, int woff, int col, int m, int k2) {
#pragma unroll
  for (int kk = 0; kk < 16; ++kk) {
    v2f a = *(const v2f*)(sm + aoff + m * HID + 4 * kk + k2);
#pragma unroll
    for (int g = 0; g < 4; ++g) {
      v2f b = *(const v2f*)(sm + woff + (g * 64 + col) * HID + 4 * kk + k2);
      acc[g] = wmma4(a, b, acc[g]);
    }
  }
}

// K=8 (padded) x-contribution: A = x tile [16][8], B = w_ih0 padded [256][8]
__device__ __forceinline__ void gemm_k8(v8f acc[4], const float* sm,
                                        int aoff, int woff, int col, int m, int k2) {
#pragma unroll
  for (int kk = 0; kk < 2; ++kk) {
    v2f a = *(const v2f*)(sm + aoff + m * 8 + 4 * kk + k2);
#pragma unroll
    for (int g = 0; g < 4; ++g) {
      v2f b = *(const v2f*)(sm + woff + (g * 64 + col) * 8 + 4 * kk + k2);
      acc[g] = wmma4(a, b, acc[g]);
    }
  }
}

__device__ __forceinline__ void lstm_act(const v8f acc[4], v8f& c, float hnew[8]) {
#pragma unroll
  for (int r = 0; r < 8; ++r) {
    float ig = sigf(acc[0][r]);
    float fg = sigf(acc[1][r]);
    float gg = tanh_f(acc[2][r]);
    float og = sigf(acc[3][r]);
    float cc = fg * c[r] + ig * gg;
    c[r] = cc;
    hnew[r] = og * tanh_f(cc);
  }
}

__device__ __forceinline__ void store_h(float* sm, int hoff, const float hnew[8],
                                        int col, int hi) {
#pragma unroll
  for (int r = 0; r < 8; ++r) sm[hoff + (r + 8 * hi) * HID + col] = hnew[r];
}

extern "C" __global__ void __launch_bounds__(NTHREADS)
lstm_fused_kernel(const float* __restrict__ x,
                  const float* __restrict__ w_ih0, const float* __restrict__ w_hh0,
                  const float* __restrict__ b_ih0, const float* __restrict__ b_hh0,
                  const float* __restrict__ w_ih1, const float* __restrict__ w_hh1,
                  const float* __restrict__ b_ih1, const float* __restrict__ b_hh1,
                  const float* __restrict__ eng_w1, const float* __restrict__ eng_b1,
                  const float* __restrict__ eng_w2, const float* __restrict__ eng_b2,
                  const float* __restrict__ prop_w1, const float* __restrict__ prop_b1,
                  const float* __restrict__ prop_w2, const float* __restrict__ prop_b2,
                  const float* __restrict__ seg_w, const float* __restrict__ seg_b,
                  float* __restrict__ out, int Btot)
{
  extern __shared__ float smem[];
  const int tid = threadIdx.x;
  const int b0  = blockIdx.x * BT;

  // ---- one-time weight load into LDS (row-major [256][64] matches memory) ----
  for (int i = tid; i < GATES * HID; i += NTHREADS) {
    smem[OFF_WHH0 + i] = w_hh0[i];
    smem[OFF_WIH1 + i] = w_ih1[i];
    smem[OFF_WHH1 + i] = w_hh1[i];
  }
  for (int i = tid; i < GATES * 8; i += NTHREADS) {
    int n = i >> 3, f = i & 7;
    smem[OFF_WIH0 + i] = (f < F_IN) ? w_ih0[n * F_IN + f] : 0.f;
  }
  for (int i = tid; i < GATES; i += NTHREADS) {
    smem[OFF_B0 + i] = b_ih0[i] + b_hh0[i];
    smem[OFF_B1 + i] = b_ih1[i] + b_hh1[i];
  }
  for (int i = tid; i < 2 * BT * HID; i += NTHREADS) smem[OFF_H0 + i] = 0.f;

  const int wid  = tid >> 5;
  const int gl   = wid >> 2;      // 0 = layer-0 group, 1 = layer-1 group (one step behind)
  const int js   = wid & 3;       // hidden slice 0..3
  const int lane = tid & 31;
  const int m    = lane & 15;     // fragment row (A) / col (B)
  const int hi   = lane >> 4;     // selects K pair within fragment
  const int k2   = 2 * hi;
  const int col  = 16 * js + m;   // gate-tile column within each 64-wide gate block

  v8f cst;                        // c0 for L0 waves, c1 for L1 waves
#pragma unroll
  for (int r = 0; r < 8; ++r) cst[r] = 0.f;

  const int sx = tid & 127;       // x staging: [16][8], 128 slots
  const int xm = sx >> 3, xf = sx & 7;
  const float* xrow = x + (size_t)(b0 + xm) * (S_LEN * F_IN) + xf;

  __syncthreads();

  for (int t = 0; t < S_LEN; ++t) {
    // P1: stage x_t (conflicting readers finished before barrier B of t-1)
    if (tid < BT * 8) smem[OFF_X + tid] = (xf < F_IN) ? xrow[t * F_IN] : 0.f;
    __syncthreads();   // A: x_t, h0[t-1], h1[t-2] all visible

    float hnew[8];
    if (gl == 0) {
      // layer 0, step t: gates = bias0 + x_t @ w_ih0^T + h0[t-1] @ w_hh0^T
      v8f acc[4];
      acc_init(acc, smem, OFF_B0, col);
      gemm_k8 (acc, smem, OFF_X,  OFF_WIH0, col, m, k2);
      gemm_k64(acc, smem, OFF_H0, OFF_WHH0, col, m, k2);
      lstm_act(acc, cst, hnew);
    } else if (t > 0) {
      // layer 1, step t-1: gates = bias1 + h0[t-1] @ w_ih1^T + h1[t-2] @ w_hh1^T
      v8f acc[4];
      acc_init(acc, smem, OFF_B1, col);
      gemm_k64(acc, smem, OFF_H0, OFF_WIH1, col, m, k2);
      gemm_k64(acc, smem, OFF_H1, OFF_WHH1, col, m, k2);
      lstm_act(acc, cst, hnew);
    }
    __syncthreads();   // B: all reads of h0[t-1] / h1[t-2] complete

    if (gl == 0)      store_h(smem, OFF_H0, hnew, col, hi);   // h0[t]
    else if (t > 0)   store_h(smem, OFF_H1, hnew, col, hi);   // h1[t-1]
    // next iteration's barrier A orders these writes before their readers
  }

  // epilogue: layer-1 group processes final step S-1 (reads h0[S-1], h1[S-2])
  __syncthreads();
  float hlast[8];
  if (gl == 1) {
    v8f acc[4];
    acc_init(acc, smem, OFF_B1, col);
    gemm_k64(acc, smem, OFF_H0, OFF_WIH1, col, m, k2);
    gemm_k64(acc, smem, OFF_H1, OFF_WHH1, col, m, k2);
    lstm_act(acc, cst, hlast);
  }
  __syncthreads();
  if (gl == 1) store_h(smem, OFF_H1, hlast, col, hi);         // h1[S-1]
  __syncthreads();

  // ======== MLP heads on last = h1[:, -1, :] (tiny; once per kernel) ========
  if (tid < BT) {
    const int b = b0 + tid;
    const float* lastp = smem + OFF_H1 + tid * HID;

    float ea = eng_b2[0];
    for (int u = 0; u < 16; ++u) {
      float s = eng_b1[u];
      for (int k = 0; k < HID; ++k) s += lastp[k] * eng_w1[u * HID + k];
      s = s > 0.f ? s : 0.f;
      ea += s * eng_w2[u];
    }
    out[b] = sigf(ea);

    float pa = prop_b2[0];
    for (int u = 0; u < 16; ++u) {
      float s = prop_b1[u];
      for (int k = 0; k < HID; ++k) s += lastp[k] * prop_w1[u * HID + k];
      s = s > 0.f ? s : 0.f;
      pa += s * prop_w2[u];
    }
    out[Btot + b] = sigf(pa);

    for (int q = 0; q < 5; ++q) {
      float s = seg_b[q];
      for (int k = 0; k < HID; ++k) s += lastp[k] * seg_w[q * HID + k];
      out[2 * Btot + b * 5 + q] = s;
    }
  }
}

extern "C" void kernel_launch(void* const* d_in, const int* in_sizes, int n_in,
                              void* d_out, int out_size, void* d_ws, size_t ws_size,
                              hipStream_t stream) {
  (void)n_in; (void)out_size; (void)d_ws; (void)ws_size;
  const float* x      = (const float*)d_in[0];
  const float* w_ih0  = (const float*)d_in[1];
  const float* w_hh0  = (const float*)d_in[2];
  const float* b_ih0  = (const float*)d_in[3];
  const float* b_hh0  = (const float*)d_in[4];
  const float* w_ih1  = (const float*)d_in[5];
  const float* w_hh1  = (const float*)d_in[6];
  const float* b_ih1  = (const float*)d_in[7];
  const float* b_hh1  = (const float*)d_in[8];
  const float* eng_w1 = (const float*)d_in[9];
  const float* eng_b1 = (const float*)d_in[10];
  const float* eng_w2 = (const float*)d_in[11];
  const float* eng_b2 = (const float*)d_in[12];
  const float* prop_w1 = (const float*)d_in[13];
  const float* prop_b1 = (const float*)d_in[14];
  const float* prop_w2 = (const float*)d_in[15];
  const float* prop_b2 = (const float*)d_in[16];
  const float* seg_w  = (const float*)d_in[17];
  const float* seg_b  = (const float*)d_in[18];
  float* out = (float*)d_out;

  const int Btot = in_sizes[0] / (S_LEN * F_IN);   // 4096
  const size_t shmem = (size_t)SMEM_FLOATS * sizeof(float); // ~210.5 KB

  hipFuncSetAttribute((const void*)lstm_fused_kernel,
                      hipFuncAttributeMaxDynamicSharedMemorySize, (int)shmem);

  dim3 grid(Btot / BT);   // 256 workgroups, one batch tile each
  dim3 block(NTHREADS);   // 8 waves: 4 layer-0 + 4 layer-1 (pipelined)
  lstm_fused_kernel<<<grid, block, shmem, stream>>>(
      x, w_ih0, w_hh0, b_ih0, b_hh0, w_ih1, w_hh1, b_ih1, b_hh1,
      eng_w1, eng_b1, eng_w2, eng_b2, prop_w1, prop_b1, prop_w2, prop_b2,
      seg_w, seg_b, out, Btot);
}